// LID_NSALoss_v2_85864986182062
// MI455X (gfx1250) — compile-verified
//
#include <hip/hip_runtime.h>
#include <hip/hip_bf16.h>
#include <math.h>

typedef __attribute__((ext_vector_type(2))) float v2f;
typedef __attribute__((ext_vector_type(8))) float v8f;
typedef __attribute__((ext_vector_type(4))) int   i32x4;

#define N_PTS   8192
#define DX      512
#define DZ      128
#define KNN     6          // K+1 = 6 kept (smallest is self)
#define EPSF    1e-07f
#define D2MIN   1e-12f

#define BLOCK_M 64         // rows per workgroup (4 waves x 16)
#define NWAVES  4
#define NTHREADS 128
#define NCOLS   32         // B chunk width (2 WMMA tiles per wave)
#define NCHUNK  (N_PTS / NCOLS)
#define LSTR    516        // padded LDS row stride (floats): 16B-aligned, conflict-free b64

// ---- optional CDNA5 async LDS staging (GLOBAL_LOAD_ASYNC_TO_LDS_B128) ----
// Probe-revealed param0 type: global (AS1) pointer to <4 x i32>, non-const.
#if __has_builtin(__builtin_amdgcn_global_load_async_to_lds_b128)
#define ASYNC_LDS 1
typedef __attribute__((address_space(1))) i32x4* gas1_b128;
typedef __attribute__((address_space(3))) i32x4* as3_b128;
#else
#define ASYNC_LDS 0
#endif

static __device__ __forceinline__ void cp16_to_lds(const float* __restrict__ g,
                                                   float* __restrict__ l) {
#if ASYNC_LDS
    __builtin_amdgcn_global_load_async_to_lds_b128(
        (gas1_b128)(float*)g, (as3_b128)l, 0, 0);
#else
    *(float4*)l = *(const float4*)g;
#endif
}

static __device__ __forceinline__ void async_wait_and_barrier() {
#if ASYNC_LDS
    asm volatile("s_wait_asynccnt 0" ::: "memory");
#endif
    __syncthreads();
}

// ---------------------------------------------------------------- row norms
__global__ void __launch_bounds__(32)
rownorm_kernel(const float* __restrict__ X, const float* __restrict__ Z,
               float* __restrict__ sqX, float* __restrict__ nX,
               float* __restrict__ sqZ, float* __restrict__ nZ) {
    const int r = blockIdx.x;
    const int l = threadIdx.x;
    const float* xr = X + (size_t)r * DX;
    float s = 0.f;
    #pragma unroll
    for (int i = l; i < DX; i += 32) { float v = xr[i]; s += v * v; }
    #pragma unroll
    for (int off = 16; off; off >>= 1) s += __shfl_xor(s, off, 32);
    if (l == 0) { sqX[r] = s; nX[r] = sqrtf(s); }

    const float* zr = Z + (size_t)r * DZ;
    float t = 0.f;
    #pragma unroll
    for (int i = l; i < DZ; i += 32) { float v = zr[i]; t += v * v; }
    #pragma unroll
    for (int off = 16; off; off >>= 1) t += __shfl_xor(t, off, 32);
    if (l == 0) { sqZ[r] = t; nZ[r] = sqrtf(t); }
}

// ------------------------------------------------- 0.98 quantile via ranking
__global__ void __launch_bounds__(256)
quantile_rank_kernel(const float* __restrict__ nX, const float* __restrict__ nZ,
                     float* __restrict__ q) {
    const int gid = blockIdx.x * blockDim.x + threadIdx.x;   // 0 .. 2*8192-1
    const int arr = gid >> 13;
    const int i   = gid & (N_PTS - 1);
    const float* a = arr ? nZ : nX;
    const float v = a[i];
    int rank = 0;
    for (int j = 0; j < N_PTS; ++j) {
        const float u = a[j];
        rank += (u < v) || (u == v && j < i);
    }
    // 0.98*(8192-1) = 8027.18 -> need sorted[8027], sorted[8028]
    if (rank == 8027) q[arr * 2 + 0] = v;
    if (rank == 8028) q[arr * 2 + 1] = v;
}

__global__ void quantile_fin_kernel(float* __restrict__ q, float* __restrict__ out) {
    const float frac = 0.98f * (float)(N_PTS - 1) - 8027.0f;   // 0.18
    q[4] = q[0] + frac * (q[1] - q[0]);   // normA1
    q[5] = q[2] + frac * (q[3] - q[2]);   // normA2
    out[0] = 0.0f;
}

// ---------------------------------------------------------------- main LID
__global__ void __launch_bounds__(NTHREADS)
lid_main_kernel(const float* __restrict__ X, const float* __restrict__ Z,
                const float* __restrict__ sqX, const float* __restrict__ sqZ,
                const float* __restrict__ q, float* __restrict__ out) {
    __shared__ float Alds[BLOCK_M * LSTR];          // 64 rows of X, ~129 KB
    __shared__ float Blds[2][NCOLS * LSTR];         // 2 x 32 cols,  ~129 KB
    __shared__ float Bsq[2][NCOLS];
    __shared__ float scratch[NWAVES][16][NCOLS + 1];

    const int tid    = threadIdx.x;                 // 0..127
    const int w      = tid >> 5;                    // wave id (0..3)
    const int lane16 = tid & 15;
    const int half   = (tid >> 4) & 1;
    const int row0   = blockIdx.x * BLOCK_M;

    const float normA1 = q[4];
    const float normA2 = q[5];

    // ---- stage the 64 A rows (once) and B chunk 0 into LDS
    for (int i = tid; i < BLOCK_M * (DX / 4); i += NTHREADS) {
        const int r  = i >> 7;          // 128 float4 per row
        const int c4 = i & 127;
        cp16_to_lds(X + (size_t)(row0 + r) * DX + c4 * 4, &Alds[r * LSTR + c4 * 4]);
    }
    for (int i = tid; i < NCOLS * (DX / 4); i += NTHREADS) {
        const int r  = i >> 7;
        const int c4 = i & 127;
        cp16_to_lds(X + (size_t)r * DX + c4 * 4, &Blds[0][r * LSTR + c4 * 4]);
    }
    if (tid < NCOLS) Bsq[0][tid] = sqX[tid];

    // per-lane top-6 state (lanes 0..15 of each wave own row w*16+lane16)
    float kd[KNN]; int kid[KNN];
    #pragma unroll
    for (int s = 0; s < KNN; ++s) { kd[s] = 3.4e38f; kid[s] = -1; }
    const int   myRow = row0 + w * 16 + lane16;
    const float mySq  = sqX[myRow];

    const float* aBase  = &Alds[(w * 16 + lane16) * LSTR + 2 * half];

    async_wait_and_barrier();

    for (int nt = 0; nt < NCHUNK; ++nt) {
        const int buf = nt & 1;

        // ---- stage next chunk into the other buffer (overlaps the WMMAs)
        if (nt + 1 < NCHUNK) {
            const size_t colRow0 = (size_t)(nt + 1) * NCOLS;
            for (int i = tid; i < NCOLS * (DX / 4); i += NTHREADS) {
                const int r  = i >> 7;
                const int c4 = i & 127;
                cp16_to_lds(X + (colRow0 + r) * DX + c4 * 4,
                            &Blds[buf ^ 1][r * LSTR + c4 * 4]);
            }
            if (tid < NCOLS) Bsq[buf ^ 1][tid] = sqX[(nt + 1) * NCOLS + tid];
        }
        if (nt + 2 < NCHUNK) {   // L2 prefetch hint (global_prefetch_b8)
            const float* p = X + ((size_t)(nt + 2) * NCOLS + (tid & 31)) * DX
                               + (tid >> 5) * 128;
            __builtin_prefetch(p, 0, 0);
        }

        // ---- C(16x32) = A(16x512) * B^T : 256 chained f32 WMMAs per wave
        const float* b0 = &Blds[buf][lane16 * LSTR + 2 * half];
        const float* b1 = &Blds[buf][(16 + lane16) * LSTR + 2 * half];
        v8f acc0 = {}, acc1 = {};
        #pragma unroll 8
        for (int k = 0; k < DX; k += 4) {
            v2f a  = *(const v2f*)(aBase + k);
            v2f f0 = *(const v2f*)(b0 + k);
            v2f f1 = *(const v2f*)(b1 + k);
            acc0 = __builtin_amdgcn_wmma_f32_16x16x4_f32(
                false, a, false, f0, (short)0, acc0, false, false);
            acc1 = __builtin_amdgcn_wmma_f32_16x16x4_f32(
                false, a, false, f1, (short)0, acc1, false, false);
        }

        // scatter C: VGPR v holds M=v (lanes 0-15) / M=v+8 (lanes 16-31)
        #pragma unroll
        for (int v = 0; v < 8; ++v) {
            scratch[w][v + 8 * half][lane16]      = acc0[v];
            scratch[w][v + 8 * half][16 + lane16] = acc1[v];
        }
        asm volatile("s_wait_dscnt 0" ::: "memory");   // wave-local LDS RAW fence

        if (half == 0) {
            const int colBase = nt * NCOLS;
            #pragma unroll 4
            for (int j = 0; j < NCOLS; ++j) {
                const float c  = scratch[w][lane16][j];
                const float d2 = mySq + Bsq[buf][j] - 2.0f * c;
                if (d2 < kd[KNN - 1]) {
                    float v = d2; int vi = colBase + j;
                    #pragma unroll
                    for (int s = 0; s < KNN; ++s) {
                        if (v < kd[s]) {
                            float td = kd[s]; int ti = kid[s];
                            kd[s] = v; kid[s] = vi; v = td; vi = ti;
                        }
                    }
                }
            }
        }
        async_wait_and_barrier();   // next buffer staged + top-k done everywhere
    }

    // ----------------------------------------------------------- tail
    if (half == 0) {
        // lid_X from the 5 NN distances (kd[0] is self)
        const float d5 = (sqrtf(fmaxf(kd[5], D2MIN)) + EPSF) / normA1;
        const float l5 = __log10f(d5);
        float lidX = 0.f;
        #pragma unroll
        for (int j = 1; j < KNN; ++j) {
            const float dj = (sqrtf(fmaxf(kd[j], D2MIN)) + EPSF) / normA1;
            lidX += __log10f(dj) - l5;
        }

        // lid_Z: distances in Z-space at the same 5 indices
        const float  szr  = sqZ[myRow];
        const float4* zr4 = (const float4*)(Z + (size_t)myRow * DZ);
        float zd[KNN - 1];
        #pragma unroll
        for (int j = 0; j < KNN - 1; ++j) {
            const int id = kid[j + 1];
            const float4* zi4 = (const float4*)(Z + (size_t)id * DZ);
            float dot = 0.f;
            #pragma unroll 8
            for (int t = 0; t < DZ / 4; ++t) {
                const float4 a = zr4[t], b = zi4[t];
                dot += a.x * b.x + a.y * b.y + a.z * b.z + a.w * b.w;
            }
            const float d2z = szr + sqZ[id] - 2.0f * dot;
            zd[j] = (sqrtf(fmaxf(d2z, D2MIN)) + EPSF) / normA2;
        }
        float mx = zd[0];
        #pragma unroll
        for (int j = 1; j < KNN - 1; ++j) mx = fmaxf(mx, zd[j]);
        const float lmx = __log10f(mx);
        float lidZ = 0.f;
        #pragma unroll
        for (int j = 0; j < KNN - 1; ++j) lidZ += __log10f(zd[j]) - lmx;

        const float diff  = 5.0f / (lidX + EPSF) - 5.0f / (lidZ + EPSF);
        const float scale = 1.0f / ((float)N_PTS * 25.0f);
        atomicAdd(out, diff * diff * scale);
    }
}

// ---------------------------------------------------------------- launch
extern "C" void kernel_launch(void* const* d_in, const int* in_sizes, int n_in,
                              void* d_out, int out_size, void* d_ws, size_t ws_size,
                              hipStream_t stream) {
    const float* X = (const float*)d_in[0];   // 8192 x 512
    const float* Z = (const float*)d_in[1];   // 8192 x 128
    float* out = (float*)d_out;

    float* ws   = (float*)d_ws;
    float* sqX  = ws;                 // 8192
    float* nX   = ws + N_PTS;         // 8192
    float* sqZ  = ws + 2 * N_PTS;     // 8192
    float* nZ   = ws + 3 * N_PTS;     // 8192
    float* q    = ws + 4 * N_PTS;     // [loX, hiX, loZ, hiZ, normA1, normA2]

    rownorm_kernel<<<N_PTS, 32, 0, stream>>>(X, Z, sqX, nX, sqZ, nZ);
    quantile_rank_kernel<<<(2 * N_PTS) / 256, 256, 0, stream>>>(nX, nZ, q);
    quantile_fin_kernel<<<1, 1, 0, stream>>>(q, out);
    lid_main_kernel<<<N_PTS / BLOCK_M, NTHREADS, 0, stream>>>(X, Z, sqX, sqZ, q, out);
}